// HGCN_84980222918800
// MI455X (gfx1250) — compile-verified
//
#include <hip/hip_runtime.h>
#include <hip/hip_bf16.h>

typedef __attribute__((ext_vector_type(2))) float v2f;
typedef __attribute__((ext_vector_type(8))) float v8f;

#define HB 4
#define HN 10000
#define F_IN 64
#define F_OUT 64
#define HT 4
#define NUM_EDGES 20000
#define NNZ 400000
#define NUM_NODES (HB * HN)            // 40000
#define ROWS (NUM_NODES * HT)          // 160000 rows of the GEMM
#define FEAT 256                       // T * F_OUT floats per node/edge
#define OUT_TOTAL (NUM_NODES * FEAT)   // 10,240,000 floats

// ---------------------------------------------------------------------------
// Kernel 1: xw[v][t][o] = sum_f x[v,f,t] * W[f,o]   via v_wmma_f32_16x16x4_f32
// One wave computes a 16-row x 64-col panel (4 accumulators, 16 K-steps).
// ---------------------------------------------------------------------------
__global__ __launch_bounds__(256) void hgcn_gemm_wmma(const float* __restrict__ x,
                                                      const float* __restrict__ W,
                                                      float* __restrict__ xw) {
    __shared__ float sW[F_IN * F_OUT];   // 16 KB of the 320 KB WGP LDS
    // cooperative 128-bit staging of W into LDS
    for (int i = threadIdx.x; i < (F_IN * F_OUT) / 4; i += 256) {
        ((float4*)sW)[i] = ((const float4*)W)[i];
    }
    __syncthreads();

    const int wave = threadIdx.x >> 5;
    const int lane = threadIdx.x & 31;
    const int tile = blockIdx.x * 8 + wave;     // 16-row tile id, 0..9999
    const int l    = lane & 15;
    const int half = lane >> 4;

    // A fragment row for this lane: m = l (same for both halves; halves differ in K)
    const int r = tile * 16 + l;                // global GEMM row = v*4 + t
    const int n = r >> 2;
    const int t = r & 3;
    const float* __restrict__ xrow = x + (size_t)n * 256 + t;   // + f*4 to index f

    v8f c0 = {}, c1 = {}, c2 = {}, c3 = {};

#pragma unroll
    for (int k = 0; k < 16; ++k) {
        const int f = k * 4 + 2 * half;         // this lane's K pair: f, f+1
        v2f a;
        a.x = xrow[f * 4];
        a.y = xrow[(f + 1) * 4];

        // B (4x16) fragment: lane half*16+l holds B[K=2*half][N=l], B[K=2*half+1][N=l]
        const float* wr0 = sW + f * F_OUT + l;
        const float* wr1 = wr0 + F_OUT;
        v2f b0, b1, b2, b3;
        b0.x = wr0[0];   b0.y = wr1[0];
        b1.x = wr0[16];  b1.y = wr1[16];
        b2.x = wr0[32];  b2.y = wr1[32];
        b3.x = wr0[48];  b3.y = wr1[48];

        c0 = __builtin_amdgcn_wmma_f32_16x16x4_f32(false, a, false, b0, (short)0, c0, false, false);
        c1 = __builtin_amdgcn_wmma_f32_16x16x4_f32(false, a, false, b1, (short)0, c1, false, false);
        c2 = __builtin_amdgcn_wmma_f32_16x16x4_f32(false, a, false, b2, (short)0, c2, false, false);
        c3 = __builtin_amdgcn_wmma_f32_16x16x4_f32(false, a, false, b3, (short)0, c3, false, false);
    }

    // D layout: VGPR j -> row (j + 8*half), col = l (per 16x16 f32 C/D table)
    float* __restrict__ outbase = xw + (size_t)(tile * 16 + 8 * half) * F_OUT + l;
#pragma unroll
    for (int j = 0; j < 8; ++j) {
        float* orow = outbase + j * F_OUT;
        orow[0]  = c0[j];
        orow[16] = c1[j];
        orow[32] = c2[j];
        orow[48] = c3[j];
    }
}

// ---------------------------------------------------------------------------
// Kernel 2: degree accumulation  D[v] += w_e ; Bdeg[e] += 1
// ---------------------------------------------------------------------------
__global__ __launch_bounds__(256) void hgcn_degrees(const long long* __restrict__ HE,
                                                    const float* __restrict__ HEWI,
                                                    float* __restrict__ D,
                                                    float* __restrict__ Bdeg) {
    int i = blockIdx.x * 256 + threadIdx.x;
    if (i >= NNZ) return;
    int v = (int)HE[i];
    int e = (int)HE[NNZ + i];
    atomicAdd(&D[v], HEWI[e]);
    atomicAdd(&Bdeg[e], 1.0f);
}

// ---------------------------------------------------------------------------
// Kernel 3: edge_feat[e] += xw[v]   (one wave per incidence; 8 floats/lane)
// ---------------------------------------------------------------------------
__global__ __launch_bounds__(256) void hgcn_edge_scatter(const long long* __restrict__ HE,
                                                         const float* __restrict__ xw,
                                                         float* __restrict__ edge_feat) {
    const int wid  = blockIdx.x * 8 + (threadIdx.x >> 5);   // nnz index, exact cover
    const int lane = threadIdx.x & 31;
    const int v = (int)HE[wid];
    const int e = (int)HE[NNZ + wid];

    const float* __restrict__ src = xw + (size_t)v * FEAT + lane * 8;
    float4 p = *(const float4*)(src);
    float4 q = *(const float4*)(src + 4);

    float* __restrict__ dst = edge_feat + (size_t)e * FEAT + lane * 8;
    atomicAdd(dst + 0, p.x); atomicAdd(dst + 1, p.y);
    atomicAdd(dst + 2, p.z); atomicAdd(dst + 3, p.w);
    atomicAdd(dst + 4, q.x); atomicAdd(dst + 5, q.y);
    atomicAdd(dst + 6, q.z); atomicAdd(dst + 7, q.w);
}

// ---------------------------------------------------------------------------
// Kernel 4: acc[v][o][t] += B_inv[e] * edge_feat[e][t][o]   (scatter into d_out,
// transposed to the output [n][o][t] layout; D_inv/bias/relu applied later)
// ---------------------------------------------------------------------------
__global__ __launch_bounds__(256) void hgcn_node_scatter(const long long* __restrict__ HE,
                                                         const float* __restrict__ edge_feat,
                                                         const float* __restrict__ Bdeg,
                                                         float* __restrict__ acc) {
    const int wid  = blockIdx.x * 8 + (threadIdx.x >> 5);
    const int lane = threadIdx.x & 31;
    const int v = (int)HE[wid];
    const int e = (int)HE[NNZ + wid];

    const float bd   = Bdeg[e];
    const float binv = (bd > 0.0f) ? (1.0f / bd) : 0.0f;

    const float* __restrict__ src = edge_feat + (size_t)e * FEAT + lane * 8;
    float4 p = *(const float4*)(src);
    float4 q = *(const float4*)(src + 4);
    float vals[8] = {p.x, p.y, p.z, p.w, q.x, q.y, q.z, q.w};

    float* __restrict__ base = acc + (size_t)v * FEAT;
    const int i0 = lane * 8;
#pragma unroll
    for (int j = 0; j < 8; ++j) {
        const int i = i0 + j;        // i = t*64 + o in edge_feat
        const int t = i >> 6;
        const int o = i & 63;
        atomicAdd(base + o * 4 + t, vals[j] * binv);
    }
}

// ---------------------------------------------------------------------------
// Kernel 5: out = relu(acc * D_inv[n] + b[o])   (in-place on d_out)
// ---------------------------------------------------------------------------
__global__ __launch_bounds__(256) void hgcn_finalize(float* __restrict__ out,
                                                     const float* __restrict__ D,
                                                     const float* __restrict__ bias) {
    int tid = blockIdx.x * 256 + threadIdx.x;     // exact cover of OUT_TOTAL
    int n = tid >> 8;
    int o = (tid >> 2) & 63;
    float d    = D[n];
    float dinv = (d > 0.0f) ? (1.0f / d) : 0.0f;
    float val  = out[tid] * dinv + bias[o];
    out[tid]   = val > 0.0f ? val : 0.0f;
}

// ---------------------------------------------------------------------------
extern "C" void kernel_launch(void* const* d_in, const int* in_sizes, int n_in,
                              void* d_out, int out_size, void* d_ws, size_t ws_size,
                              hipStream_t stream) {
    const float*     x    = (const float*)d_in[0];
    const long long* HE   = (const long long*)d_in[1];
    const float*     HEWI = (const float*)d_in[2];
    const float*     W    = (const float*)d_in[3];
    const float*     bias = (const float*)d_in[4];
    float* out = (float*)d_out;

    // workspace layout
    char* ws = (char*)d_ws;
    float* xw        = (float*)(ws);                                   // 40,960,000 B
    float* edge_feat = (float*)(ws + (size_t)ROWS * F_OUT * 4);        // 20,480,000 B
    float* Dv        = (float*)(ws + (size_t)(ROWS * F_OUT + NUM_EDGES * FEAT) * 4);
    float* Bdeg      = (float*)(ws + (size_t)(ROWS * F_OUT + NUM_EDGES * FEAT + NUM_NODES) * 4);

    // zero all accumulators (edge_feat, D, Bdeg are contiguous) and d_out
    size_t accum_bytes = (size_t)(NUM_EDGES * FEAT + NUM_NODES + NUM_EDGES) * 4;
    hipMemsetAsync(edge_feat, 0, accum_bytes, stream);
    hipMemsetAsync(out, 0, (size_t)out_size * 4, stream);

    // 1) GEMM via f32 WMMA: 10000 row tiles, 8 waves/block
    hgcn_gemm_wmma<<<ROWS / 16 / 8, 256, 0, stream>>>(x, W, xw);
    // 2) degrees
    hgcn_degrees<<<(NNZ + 255) / 256, 256, 0, stream>>>(HE, HEWI, Dv, Bdeg);
    // 3) node -> hyperedge aggregation
    hgcn_edge_scatter<<<NNZ / 8, 256, 0, stream>>>(HE, xw, edge_feat);
    // 4) hyperedge -> node aggregation (into d_out, transposed)
    hgcn_node_scatter<<<NNZ / 8, 256, 0, stream>>>(HE, edge_feat, Bdeg, out);
    // 5) D_inv scale + bias + relu
    hgcn_finalize<<<OUT_TOTAL / 256, 256, 0, stream>>>(out, Dv, bias);
}